// EnhancedMultiScaleAdaptiveElasticityLossWithLame_29703993819927
// MI455X (gfx1250) — compile-verified
//
#include <hip/hip_runtime.h>

typedef float v2f __attribute__((ext_vector_type(2)));
typedef float v8f __attribute__((ext_vector_type(8)));

namespace {
constexpr int       kD0 = 160;
constexpr int       kB  = 2;
constexpr long long kN3Full = (long long)kD0 * kD0 * kD0;   // 4,096,000
constexpr long long kHalf3  = (long long)80 * 80 * 80;      // 512,000
// workspace layout (float offsets)
constexpr long long kOffIG   = 0;
constexpr long long kOffTA   = kOffIG + kB * kN3Full;       //  8,192,000
constexpr long long kOffTB   = kOffTA + kB * kN3Full;       // 16,384,000
constexpr long long kOffImgS = kOffTB + kB * kN3Full;       // 24,576,000
constexpr long long kOffDefS = kOffImgS + kB * kHalf3;      // 25,600,000
constexpr long long kOffPart = kOffDefS + kB * 3 * kHalf3;  // 28,672,000

// Normalized 5-tap Gaussian, sigma=1.1 (k[t] = exp(-(t-2)^2/2.42)/sum), f32-rounded
constexpr float kG0 = 0.07076642f;   // |d-2| == 2
constexpr float kG1 = 0.24446039f;   // |d-2| == 1
constexpr float kG2 = 0.36954639f;   // |d-2| == 0
}

// ---------------------------------------------------------------- helpers
// Banded A-matrix weight: kern[d] for d in [0,4], else 0.  (|d-2|<=2 <=> 0<=d<=4)
__device__ __forceinline__ float bandw(int d) {
  int ad = abs(d - 2);
  float w = (ad == 0) ? kG2 : ((ad == 1) ? kG1 : kG0);
  return (ad <= 2) ? w : 0.f;
}

__device__ __forceinline__ int reflect_idx(int i, int n) {
  if (i < 0) return -i;
  if (i > n - 1) return 2 * (n - 1) - i;
  return i;
}

// jnp.gradient along one axis (stride str), one-sided at boundaries
__device__ __forceinline__ float grad1(const float* __restrict__ p, long long s,
                                       int i, int n, long long str) {
  if (i == 0)     return p[s + str] - p[s];
  if (i == n - 1) return p[s] - p[s - str];
  return 0.5f * (p[s + str] - p[s - str]);
}

// ------------------------------------------------ trilinear resize (align_corners)
__global__ __launch_bounds__(256) void resize_tri(const float* __restrict__ src,
                                                  float* __restrict__ dst,
                                                  int nin, int nout, long long count) {
  long long i = (long long)blockIdx.x * blockDim.x + threadIdx.x;
  if (i >= count) return;
  int z = (int)(i % nout); long long t = i / nout;
  int y = (int)(t % nout); t /= nout;
  int x = (int)(t % nout); int bc = (int)(t / nout);

  float r  = (float)(nin - 1) / (float)(nout - 1);
  float cx = x * r, cy = y * r, cz = z * r;
  int ix = (int)floorf(cx), iy = (int)floorf(cy), iz = (int)floorf(cz);
  float fx = cx - ix, fy = cy - iy, fz = cz - iz;
  int ix1 = min(ix + 1, nin - 1), iy1 = min(iy + 1, nin - 1), iz1 = min(iz + 1, nin - 1);

  const float* sv = src + (long long)bc * nin * nin * nin;
  auto at = [&](int a, int b, int c) -> float {
    return sv[((long long)a * nin + b) * nin + c];
  };
  // X interp first, then Y, then Z (matches reference separable order)
  float c00 = at(ix, iy,  iz ) * (1.f - fx) + at(ix1, iy,  iz ) * fx;
  float c10 = at(ix, iy1, iz ) * (1.f - fx) + at(ix1, iy1, iz ) * fx;
  float c01 = at(ix, iy,  iz1) * (1.f - fx) + at(ix1, iy,  iz1) * fx;
  float c11 = at(ix, iy1, iz1) * (1.f - fx) + at(ix1, iy1, iz1) * fx;
  float c0  = c00 * (1.f - fy) + c10 * fy;
  float c1  = c01 * (1.f - fy) + c11 * fy;
  dst[i] = c0 * (1.f - fz) + c1 * fz;
}

// ------------------------------------------------ image gradient magnitude
__global__ __launch_bounds__(256) void gradmag_kernel(const float* __restrict__ img,
                                                      float* __restrict__ ig,
                                                      int n, long long total) {
  long long i = (long long)blockIdx.x * blockDim.x + threadIdx.x;
  if (i >= total) return;
  long long n3 = (long long)n * n * n;
  int z = (int)(i % n); long long t = i / n;
  int y = (int)(t % n); t /= n;
  int x = (int)(t % n); int b = (int)(t / n);
  const float* v = img + (long long)b * n3;
  long long s = ((long long)x * n + y) * (long long)n + z;
  float gx = grad1(v, s, x, n, (long long)n * n);
  float gy = grad1(v, s, y, n, n);
  float gz = grad1(v, s, z, n, 1);
  ig[i] = sqrtf(gx * gx + gy * gy + gz * gz);
}

// ------------------------------------------------ blur along X and Y (VALU, coalesced in z)
__global__ __launch_bounds__(256) void blur_x_kernel(const float* __restrict__ src,
                                                     float* __restrict__ dst,
                                                     int n, long long total) {
  long long i = (long long)blockIdx.x * blockDim.x + threadIdx.x;
  if (i >= total) return;
  const float k[5] = {kG0, kG1, kG2, kG1, kG0};
  long long n3 = (long long)n * n * n;
  int z = (int)(i % n); long long t = i / n;
  int y = (int)(t % n); t /= n;
  int x = (int)(t % n); int b = (int)(t / n);
  const float* v = src + (long long)b * n3;
  float acc = 0.f;
#pragma unroll
  for (int tt = 0; tt < 5; ++tt) {
    int xi = reflect_idx(x - 2 + tt, n);
    acc += k[tt] * v[((long long)xi * n + y) * (long long)n + z];
  }
  dst[i] = acc;
}

__global__ __launch_bounds__(256) void blur_y_kernel(const float* __restrict__ src,
                                                     float* __restrict__ dst,
                                                     int n, long long total) {
  long long i = (long long)blockIdx.x * blockDim.x + threadIdx.x;
  if (i >= total) return;
  const float k[5] = {kG0, kG1, kG2, kG1, kG0};
  long long n3 = (long long)n * n * n;
  int z = (int)(i % n); long long t = i / n;
  int y = (int)(t % n); t /= n;
  int x = (int)(t % n); int b = (int)(t / n);
  const float* v = src + (long long)b * n3;
  float acc = 0.f;
#pragma unroll
  for (int tt = 0; tt < 5; ++tt) {
    int yi = reflect_idx(y - 2 + tt, n);
    acc += k[tt] * v[((long long)x * n + yi) * (long long)n + z];
  }
  dst[i] = acc;
}

// ------------------------------------------------ blur along Z via WMMA f32 16x16x4
// One wave = one 16(z-out) x 16(y-lines) tile; D = A(16x20 banded weights) * B(20x16 data),
// accumulated as 5 chained V_WMMA_F32_16X16X4_F32 ops. Input patch staged into LDS with
// GLOBAL_LOAD_ASYNC_TO_LDS_B32 (ASYNCcnt) — builtin confirmed present on this toolchain.
__global__ __launch_bounds__(128) void blur_z_wmma_kernel(const float* __restrict__ src,
                                                          float* __restrict__ dst,
                                                          int n, int B) {
  __shared__ float ldsIn[4][320];   // [wave][16 lines * 20 z] (stride 20 -> bank-conflict-free)
  __shared__ float ldsOut[4][256];  // [wave][16 z * 16 y] transpose staging

  const int lane = threadIdx.x & 31;
  const int wv   = threadIdx.x >> 5;

  const int YT = (n + 15) >> 4;
  const int ZT = YT;
  const int total = B * n * YT * ZT;
  int tile = blockIdx.x * 4 + wv;
  const bool active = tile < total;
  if (!active) tile = 0;  // inactive waves process tile 0, skip stores (keep syncthreads legal)

  int zt = tile % ZT; int t2 = tile / ZT;
  int yt = t2 % YT;   t2 /= YT;
  int x  = t2 % n;    int b = t2 / n;
  const int y0 = yt * 16, z0 = zt * 16;

  const float* plane = src + ((long long)b * n + x) * (long long)n * n;

  // stage 16x20 input patch (reflect-padded in z, clamped in y); uniform trip count
#if __has_builtin(__builtin_amdgcn_global_load_async_to_lds_b32) && \
    __has_builtin(__builtin_amdgcn_s_wait_asynccnt)
  {
    auto* ldsBase = (__attribute__((address_space(3))) int*)&ldsIn[wv][0];
    for (int i = lane; i < 320; i += 32) {
      int yy = i / 20, t = i - yy * 20;
      int y = y0 + yy; if (y > n - 1) y = n - 1;
      int zi = reflect_idx(z0 - 2 + t, n);
      __builtin_amdgcn_global_load_async_to_lds_b32(
          (__attribute__((address_space(1))) int*)(plane + (long long)y * n + zi),
          ldsBase + i,
          /*offset=*/0, /*cpol=*/0);
    }
    __builtin_amdgcn_s_wait_asynccnt(0);
  }
#else
  for (int i = lane; i < 320; i += 32) {
    int yy = i / 20, t = i - yy * 20;
    int y = y0 + yy; if (y > n - 1) y = n - 1;
    int zi = reflect_idx(z0 - 2 + t, n);
    ldsIn[wv][i] = plane[(long long)y * n + zi];
  }
#endif
  __syncthreads();

  const int m    = lane & 15;           // A row / B column
  const int koff = (lane < 16) ? 0 : 2; // K sub-slot per ISA f32 A/B fragment layout
  v8f c = {0.f, 0.f, 0.f, 0.f, 0.f, 0.f, 0.f, 0.f};
#pragma unroll
  for (int g = 0; g < 5; ++g) {
    const int k0 = g * 4 + koff;
    v2f a, bf;
    a.x = bandw(k0 - m);                // banded weights: A[m,j'] = kern[j'-m]
    a.y = bandw(k0 + 1 - m);
    bf.x = ldsIn[wv][m * 20 + k0];
    bf.y = ldsIn[wv][m * 20 + k0 + 1];
    c = __builtin_amdgcn_wmma_f32_16x16x4_f32(false, a, false, bf, (short)0, c, false, false);
  }

  // D layout: VGPR r -> rows (r, r+8); write [z][y] tile to LDS, then store coalesced in z
  const int mbase = (lane < 16) ? 0 : 8;
#pragma unroll
  for (int r = 0; r < 8; ++r) ldsOut[wv][(mbase + r) * 16 + m] = c[r];
  __syncthreads();

  float* oplane = dst + ((long long)b * n + x) * (long long)n * n;
  for (int i = lane; i < 256; i += 32) {
    int yy = i >> 4, zz = i & 15;
    int y = y0 + yy, z = z0 + zz;
    if (active && y < n && z < n) oplane[(long long)y * n + z] = ldsOut[wv][zz * 16 + yy];
  }
}

// ------------------------------------------------ strain energy + block reduction
__global__ __launch_bounds__(256) void energy_reduce_kernel(const float* __restrict__ def,
                                                            const float* __restrict__ igsm,
                                                            float* __restrict__ partials,
                                                            int n, long long n3, int B) {
  __shared__ float red[256];
  const long long total  = (long long)B * n3;
  const long long stride = (long long)gridDim.x * blockDim.x;
  float acc = 0.f;
  for (long long i = (long long)blockIdx.x * blockDim.x + threadIdx.x; i < total; i += stride) {
    int z = (int)(i % n); long long t = i / n;
    int y = (int)(t % n); t /= n;
    int x = (int)(t % n); int b = (int)(t / n);
    long long s = ((long long)x * n + y) * (long long)n + z;
    const float* u  = def + ((long long)b * 3 + 0) * n3;
    const float* v  = def + ((long long)b * 3 + 1) * n3;
    const float* w2 = def + ((long long)b * 3 + 2) * n3;
    __builtin_prefetch(&igsm[i + stride], 0, 1);   // global_prefetch_b8

    const long long sx = (long long)n * n;
    float dudx = grad1(u,  s, x, n, sx), dudy = grad1(u,  s, y, n, n), dudz = grad1(u,  s, z, n, 1);
    float dvdx = grad1(v,  s, x, n, sx), dvdy = grad1(v,  s, y, n, n), dvdz = grad1(v,  s, z, n, 1);
    float dwdx = grad1(w2, s, x, n, sx), dwdy = grad1(w2, s, y, n, n), dwdz = grad1(w2, s, z, n, 1);

    float g   = igsm[i];
    float lam = fminf(fmaxf(1.0f + 2.0f * g, 0.1f), 10.0f);
    float mu  = fminf(fmaxf(0.5f + 1.0f * g, 0.1f), 10.0f);

    float Exx = dudx, Eyy = dvdy, Ezz = dwdz;
    float Exy = 0.5f * (dudy + dvdx);
    float Exz = 0.5f * (dudz + dwdx);
    float Eyz = 0.5f * (dvdz + dwdy);
    float tr  = Exx + Eyy + Ezz;
    float e   = 0.5f * lam * tr * tr +
                mu * (Exx * Exx + Eyy * Eyy + Ezz * Ezz +
                      2.0f * (Exy * Exy + Exz * Exz + Eyz * Eyz));
    acc += (1.0f + 5.0f * g) * e;
  }
  red[threadIdx.x] = acc;
  __syncthreads();
  for (int o = 128; o > 0; o >>= 1) {
    if ((int)threadIdx.x < o) red[threadIdx.x] += red[threadIdx.x + o];
    __syncthreads();
  }
  if (threadIdx.x == 0) partials[blockIdx.x] = red[0];
}

// ------------------------------------------------ final deterministic combine + Jacobian penalty
__global__ void finalize_kernel(const float* __restrict__ part,
                                const float* __restrict__ def,
                                float* __restrict__ out) {
  if (threadIdx.x != 0 || blockIdx.x != 0) return;
  const int   cnt[3]  = {2048, 512, 128};
  const int   off[3]  = {0, 2048, 2560};
  const float sw[3]   = {1.0f, 0.5f, 0.25f};
  const int   ns[3]   = {160, 80, 40};
  float total = 0.f;
  for (int s = 0; s < 3; ++s) {
    float sum = 0.f;
    for (int j = 0; j < cnt[s]; ++j) sum += part[off[s] + j];
    float inv = 1.0f / ((float)kB * (float)ns[s] * (float)ns[s] * (float)ns[s]);
    total += sw[s] * sum * inv;
  }
  // Jacobian penalty at center voxel of full-res field
  const int n = 160, c = 80;
  auto D = [&](int b, int comp, int x, int y, int z) -> float {
    return def[((((long long)b * 3 + comp) * n + x) * n + y) * (long long)n + z];
  };
  float pen = 0.f;
  for (int b = 0; b < kB; ++b) {
    float J[3][3];
    for (int comp = 0; comp < 3; ++comp) {
      J[comp][0] = 0.5f * (D(b, comp, c + 1, c, c) - D(b, comp, c - 1, c, c));
      J[comp][1] = 0.5f * (D(b, comp, c, c + 1, c) - D(b, comp, c, c - 1, c));
      J[comp][2] = 0.5f * (D(b, comp, c, c, c + 1) - D(b, comp, c, c, c - 1));
    }
    float det = J[0][0] * (J[1][1] * J[2][2] - J[1][2] * J[2][1])
              - J[0][1] * (J[1][0] * J[2][2] - J[1][2] * J[2][0])
              + J[0][2] * (J[1][0] * J[2][1] - J[1][1] * J[2][0]);
    pen += fmaxf(-det, 0.f);
  }
  total += 0.1f * (pen / (float)kB);
  out[0] = total;
}

// ------------------------------------------------ host-side orchestration
extern "C" void kernel_launch(void* const* d_in, const int* in_sizes, int n_in,
                              void* d_out, int out_size, void* d_ws, size_t ws_size,
                              hipStream_t stream) {
  (void)in_sizes; (void)n_in; (void)out_size; (void)ws_size;
  const float* def_in = (const float*)d_in[0];
  const float* img_in = (const float*)d_in[1];
  float* ws   = (float*)d_ws;
  float* IG   = ws + kOffIG;
  float* TA   = ws + kOffTA;
  float* TB   = ws + kOffTB;
  float* IMGS = ws + kOffImgS;
  float* DEFS = ws + kOffDefS;
  float* PART = ws + kOffPart;

  const int enBlocks[3] = {2048, 512, 128};
  const int partOff[3]  = {0, 2048, 2560};

  for (int s = 0; s < 3; ++s) {
    const int n = kD0 >> s;
    const long long n3  = (long long)n * n * n;
    const long long tot = (long long)kB * n3;
    const float* img_s = img_in;
    const float* def_s = def_in;
    if (s > 0) {
      const long long cI = (long long)kB * n3;
      resize_tri<<<(int)((cI + 255) / 256), 256, 0, stream>>>(img_in, IMGS, kD0, n, cI);
      const long long cD = (long long)kB * 3 * n3;
      resize_tri<<<(int)((cD + 255) / 256), 256, 0, stream>>>(def_in, DEFS, kD0, n, cD);
      img_s = IMGS; def_s = DEFS;
    }
    gradmag_kernel<<<(int)((tot + 255) / 256), 256, 0, stream>>>(img_s, IG, n, tot);
    blur_x_kernel <<<(int)((tot + 255) / 256), 256, 0, stream>>>(IG, TA, n, tot);
    blur_y_kernel <<<(int)((tot + 255) / 256), 256, 0, stream>>>(TA, TB, n, tot);
    const int YT = (n + 15) / 16;
    const int T  = kB * n * YT * YT;
    blur_z_wmma_kernel<<<(T + 3) / 4, 128, 0, stream>>>(TB, IG, n, kB);
    energy_reduce_kernel<<<enBlocks[s], 256, 0, stream>>>(def_s, IG, PART + partOff[s],
                                                          n, n3, kB);
  }
  finalize_kernel<<<1, 32, 0, stream>>>(PART, def_in, (float*)d_out);
}